// GCNEncoder_89635967467596
// MI455X (gfx1250) — compile-verified
//
#include <hip/hip_runtime.h>

typedef float v2f __attribute__((ext_vector_type(2)));
typedef float v8f __attribute__((ext_vector_type(8)));

// ---------------------------------------------------------------------------
// Degree / normalization
// ---------------------------------------------------------------------------
__global__ void zero_f32_kernel(float* __restrict__ p, int n) {
    int i = blockIdx.x * blockDim.x + threadIdx.x;
    if (i < n) p[i] = 0.0f;
}

__global__ void degree_accum_kernel(const long long* __restrict__ dst,
                                    float* __restrict__ deg, int E) {
    int i = blockIdx.x * blockDim.x + threadIdx.x;
    if (i < E) atomicAdd(&deg[dst[i]], 1.0f);
}

// dis[i] = 1/sqrt(deg[i] + 1)   (+1 = self loop; always > 0)
__global__ void deg_to_dis_kernel(float* __restrict__ d, int n) {
    int i = blockIdx.x * blockDim.x + threadIdx.x;
    if (i < n) d[i] = 1.0f / sqrtf(d[i] + 1.0f);
}

// ---------------------------------------------------------------------------
// fp32 WMMA GEMM: C[M,N] = op(A[M,K]) @ B[K,N], one wave per 16x16 tile.
// op = ReLU on A when RELU=true (fuses relu(agg1) into layer-2 transform).
// Uses V_WMMA_F32_16X16X4_F32 (full fp32, matches reference precision class).
// ---------------------------------------------------------------------------
template <bool RELU>
__global__ void gemm16_wmma_kernel(const float* __restrict__ A,
                                   const float* __restrict__ B,
                                   float* __restrict__ C,
                                   int M, int K, int N) {
    const int lane = threadIdx.x & 31;
    const int wave = (blockIdx.x * blockDim.x + threadIdx.x) >> 5;
    const int ntiles = N >> 4;
    const int mtiles = M >> 4;
    if (wave >= ntiles * mtiles) return;      // wave-uniform: EXEC stays all-1s
    const int tm = wave / ntiles;
    const int tn = wave - tm * ntiles;

    const int nIdx = lane & 15;   // M-row for A frag, N-col for B/C frags
    const int half = lane >> 4;

    // A-frag: lane (m=nIdx, half) holds A[m][k + 2*half + {0,1}]
    const float* aptr = A + (size_t)(tm * 16 + nIdx) * K + 2 * half;
    // B-frag: lane (n=nIdx, half) holds B[k + 2*half + {0,1}][n]
    const float* bptr = B + (size_t)(2 * half) * N + tn * 16 + nIdx;

    v8f acc = {};
    for (int k = 0; k < K; k += 4) {
        v2f a, b;
        a.x = aptr[k];
        a.y = aptr[k + 1];
        if (RELU) { a.x = fmaxf(a.x, 0.0f); a.y = fmaxf(a.y, 0.0f); }
        b.x = bptr[(size_t)k * N];
        b.y = bptr[(size_t)(k + 1) * N];
        acc = __builtin_amdgcn_wmma_f32_16x16x4_f32(
            /*neg_a=*/false, a, /*neg_b=*/false, b,
            /*c_mod=*/(short)0, acc, /*reuse_a=*/false, /*reuse_b=*/false);
    }

    // C/D layout: VGPR v -> row (v + 8*half), col nIdx
    float* cptr = C + (size_t)(tm * 16 + 8 * half) * N + tn * 16 + nIdx;
#pragma unroll
    for (int v = 0; v < 8; ++v) cptr[(size_t)v * N] = acc[v];
}

// ---------------------------------------------------------------------------
// agg[i,c] = bias[c] + dis[i]^2 * h[i,c]
// (self-loop message + bias folded into accumulator init -> no atomics needed)
// ---------------------------------------------------------------------------
__global__ void init_selfloop_bias_kernel(const float* __restrict__ h,
                                          const float* __restrict__ dis,
                                          const float* __restrict__ bias,
                                          float* __restrict__ agg,
                                          int total, int F) {
    int i = blockIdx.x * blockDim.x + threadIdx.x;
    if (i >= total) return;
    int node = i / F;
    int c = i - node * F;
    float w = dis[node];
    w *= w;
    agg[i] = bias[c] + w * h[i];
}

// ---------------------------------------------------------------------------
// Edge scatter: agg[dst] += dis[src]*dis[dst] * h[src]
// One thread handles 4 channels of one edge (float4 load, 4 fp32 atomics).
// Consecutive threads cover consecutive channels of the same edge -> coalesced.
// h/agg fit in the 192 MB L2, so atomics resolve in L2, not HBM.
// ---------------------------------------------------------------------------
__global__ void scatter_edges_kernel(const long long* __restrict__ src,
                                     const long long* __restrict__ dst,
                                     const float* __restrict__ dis,
                                     const float* __restrict__ h,
                                     float* __restrict__ agg,
                                     int E, int F) {
    int idx = blockIdx.x * blockDim.x + threadIdx.x;
    int vec = F >> 2;
    int e = idx / vec;
    if (e >= E) return;
    int c = (idx - e * vec) << 2;

    long long s = src[e];
    long long d = dst[e];
    float w = dis[s] * dis[d];

    const float4 hv = *reinterpret_cast<const float4*>(h + (size_t)s * F + c);
    float* o = agg + (size_t)d * F + c;
    atomicAdd(o + 0, w * hv.x);
    atomicAdd(o + 1, w * hv.y);
    atomicAdd(o + 2, w * hv.z);
    atomicAdd(o + 3, w * hv.w);
}

// ---------------------------------------------------------------------------
// Launch
// ---------------------------------------------------------------------------
extern "C" void kernel_launch(void* const* d_in, const int* in_sizes, int n_in,
                              void* d_out, int out_size, void* d_ws, size_t ws_size,
                              hipStream_t stream) {
    const float* x      = (const float*)d_in[0];
    const long long* ei = (const long long*)d_in[1];   // int64 edge_index [2,E]
    const float* W1     = (const float*)d_in[2];
    const float* b1     = (const float*)d_in[3];
    const float* W2     = (const float*)d_in[4];
    const float* b2     = (const float*)d_in[5];
    float* out          = (float*)d_out;

    const int hid  = in_sizes[3];            // 128
    const int outc = in_sizes[5];            // 64
    const int inc  = in_sizes[2] / hid;      // 256
    const int Nn   = in_sizes[0] / inc;      // 50000
    const int E    = in_sizes[1] / 2;        // 800000
    const long long* src = ei;
    const long long* dst = ei + E;

    // workspace carve-out (floats): dis | h1 | agg1 | h2   (~64.2 MB total)
    float* ws   = (float*)d_ws;
    size_t off  = ((size_t)Nn + 63) & ~(size_t)63;
    float* dis  = ws;
    float* h1   = ws + off;
    float* agg1 = h1 + (size_t)Nn * hid;
    float* h2   = agg1 + (size_t)Nn * hid;

    const int B = 256;   // 8 wave32 waves per block

    // --- normalization: deg -> dis = 1/sqrt(deg+1) ---
    zero_f32_kernel<<<(Nn + B - 1) / B, B, 0, stream>>>(dis, Nn);
    degree_accum_kernel<<<(E + B - 1) / B, B, 0, stream>>>(dst, dis, E);
    deg_to_dis_kernel<<<(Nn + B - 1) / B, B, 0, stream>>>(dis, Nn);

    // --- layer 1: h1 = x @ W1 (WMMA fp32) ---
    {
        int tiles  = (Nn / 16) * (hid / 16);
        int blocks = (tiles + (B / 32) - 1) / (B / 32);
        gemm16_wmma_kernel<false><<<blocks, B, 0, stream>>>(x, W1, h1, Nn, inc, hid);
    }
    // agg1 = b1 + dis^2 * h1  (self loop + bias)
    init_selfloop_bias_kernel<<<((Nn * hid) + B - 1) / B, B, 0, stream>>>(
        h1, dis, b1, agg1, Nn * hid, hid);
    // agg1[dst] += norm * h1[src]
    scatter_edges_kernel<<<(E * (hid / 4) + B - 1) / B, B, 0, stream>>>(
        src, dst, dis, h1, agg1, E, hid);

    // --- layer 2: h2 = relu(agg1) @ W2 (WMMA fp32, fused ReLU) ---
    {
        int tiles  = (Nn / 16) * (outc / 16);
        int blocks = (tiles + (B / 32) - 1) / (B / 32);
        gemm16_wmma_kernel<true><<<blocks, B, 0, stream>>>(agg1, W2, h2, Nn, hid, outc);
    }
    // out = b2 + dis^2 * h2
    init_selfloop_bias_kernel<<<((Nn * outc) + B - 1) / B, B, 0, stream>>>(
        h2, dis, b2, out, Nn * outc, outc);
    // out[dst] += norm * h2[src]
    scatter_edges_kernel<<<(E * (outc / 4) + B - 1) / B, B, 0, stream>>>(
        src, dst, dis, h2, out, E, outc);
}